// OriginNoiseBilateral_50113678410597
// MI455X (gfx1250) — compile-verified
//
#include <hip/hip_runtime.h>

#define TILE   32
#define RMAX   8
#define HALO   (TILE + 2*RMAX)   /* 48 */
#define IMG_H  512
#define IMG_W  512
#define IMG_HW (IMG_H * IMG_W)
#define LOG2E  1.4426950408889634f

#if defined(__has_builtin)
#if __has_builtin(__builtin_amdgcn_global_load_async_to_lds_b32) && \
    __has_builtin(__builtin_amdgcn_s_wait_asynccnt)
#define USE_ASYNC_LDS 1
#endif
#if __has_builtin(__builtin_amdgcn_exp2f)
#define FAST_EXP2(x) __builtin_amdgcn_exp2f(x)   /* bare v_exp_f32 */
#endif
#endif
#ifndef FAST_EXP2
#define FAST_EXP2(x) __expf((x) * 0.6931471805599453f)
#endif

typedef __attribute__((address_space(1))) int global_i32;
typedef __attribute__((address_space(3))) int local_i32;

__global__ __launch_bounds__(256, 2)
void bilateral_kernel(const float* __restrict__ img,
                      const float* __restrict__ params,
                      float* __restrict__ out)
{
    // [row][col] of float4: channels interleaved -> one ds_load_b128 per tap.
    __shared__ float4 tile4[HALO * HALO];   // 36,864 bytes
    float* ldsf = (float*)tile4;

    const int tid = threadIdx.x;
    const int tx = blockIdx.x, ty = blockIdx.y, n = blockIdx.z;
    const int h0 = ty * TILE - RMAX;
    const int w0 = tx * TILE - RMAX;

    const float* __restrict__ planes = img + (size_t)n * 3 * IMG_HW;

    // ---- per-image scalar params (uniform across the block) ----
    const float p0 = params[n * 3 + 0];
    const float p1 = params[n * 3 + 1];
    const float p2 = params[n * 3 + 2];
    int radius = (int)p0 * 7 + 1;          // window = int(p0)*14+3 ; r = (w-1)/2
    radius = radius > RMAX ? RMAX : (radius < 0 ? 0 : radius);
    const float sc = p1 * 99.0f + 1.0f;
    const float ss = p2 * 99.0f + 1.0f;
    // exp(a) == exp2(a*log2e): fold log2e + the 255^2 image scale into the
    // two exponent constants so each tap is fma + bare v_exp_f32.
    const float neg_inv2c = -LOG2E * (255.0f * 255.0f) / (2.0f * sc * sc);
    const float neg_inv2s = -LOG2E / (2.0f * ss * ss);

    // ---- stage 48x48x3 halo into LDS (edge-clamped), async on gfx1250 ----
    // 48*48 = 2304 = 9*256 spatial slots per channel.
    #pragma unroll
    for (int c = 0; c < 3; ++c) {
        const float* __restrict__ cplane = planes + c * IMG_HW;
        #pragma unroll
        for (int j = 0; j < 9; ++j) {
            const int s  = tid + j * 256;
            const int ry = s / HALO;
            const int rx = s - ry * HALO;
            int gy = h0 + ry; gy = gy < 0 ? 0 : (gy > IMG_H - 1 ? IMG_H - 1 : gy);
            int gx = w0 + rx; gx = gx < 0 ? 0 : (gx > IMG_W - 1 ? IMG_W - 1 : gx);
            float* src = const_cast<float*>(cplane + (size_t)gy * IMG_W + gx);
            float* dst = ldsf + (s * 4 + c);
#ifdef USE_ASYNC_LDS
            __builtin_amdgcn_global_load_async_to_lds_b32(
                (global_i32*)src, (local_i32*)dst, 0, 0);
#else
            *dst = *src;
#endif
        }
    }
#ifdef USE_ASYNC_LDS
    __builtin_amdgcn_s_wait_asynccnt(0);
#endif
    __syncthreads();

    // ---- each thread owns 4 pixels: same lx, rows ly0 + {0,8,16,24} ----
    const int lx  = tid & 31;        // 0..31
    const int ly0 = tid >> 5;        // 0..7

    float x0[4], x1[4], x2[4];
    float num0[4], num1[4], num2[4], den[4];
    #pragma unroll
    for (int k = 0; k < 4; ++k) {
        const int ly = ly0 + k * 8;
        const float4 v = tile4[(ly + RMAX) * HALO + (lx + RMAX)];
        x0[k] = v.x; x1[k] = v.y; x2[k] = v.z;
        num0[k] = 0.0f; num1[k] = 0.0f; num2[k] = 0.0f; den[k] = 0.0f;
    }

    // One bilateral tap for thread-pixel k at LDS float4 index `idx`.
    // `sp` is the (already log2e-scaled) spatial term.
    auto tap = [&](int k, int idx, float sp) {
        const float4 s4 = tile4[idx];
        const float d0 = s4.x - x0[k];
        const float d1 = s4.y - x1[k];
        const float d2 = s4.z - x2[k];
        const float dd = fmaf(d0, d0, fmaf(d1, d1, d2 * d2));
        const float w  = FAST_EXP2(fmaf(dd, neg_inv2c, sp));
        num0[k] = fmaf(w, s4.x, num0[k]);
        num1[k] = fmaf(w, s4.y, num1[k]);
        num2[k] = fmaf(w, s4.z, num2[k]);
        den[k] += w;
    };

    if (radius == RMAX) {
        // ---- fast path (actual inputs): 17x17, dx fully unrolled ----
        for (int dy = -RMAX; dy <= RMAX; ++dy) {
            const int rowb = (ly0 + RMAX + dy) * HALO + RMAX + lx;
            const float spy = (float)(dy * dy) * neg_inv2s;
            #pragma unroll
            for (int dx = -RMAX; dx <= RMAX; ++dx) {
                const float sp = fmaf((float)(dx * dx), neg_inv2s, spy);
                #pragma unroll
                for (int k = 0; k < 4; ++k)
                    tap(k, rowb + k * (8 * HALO) + dx, sp);
            }
        }
    } else {
        // ---- generic path: any radius in [0,8] (block-uniform bounds) ----
        for (int dy = -radius; dy <= radius; ++dy) {
            const int rowb = (ly0 + RMAX + dy) * HALO + RMAX + lx;
            const float spy = (float)(dy * dy) * neg_inv2s;
            for (int dx = -radius; dx <= radius; ++dx) {
                const float sp = fmaf((float)(dx * dx), neg_inv2s, spy);
                #pragma unroll
                for (int k = 0; k < 4; ++k)
                    tap(k, rowb + k * (8 * HALO) + dx, sp);
            }
        }
    }

    // ---- write NCHW output (coalesced within each wave) ----
    float* __restrict__ outn = out + (size_t)n * 3 * IMG_HW;
    #pragma unroll
    for (int k = 0; k < 4; ++k) {
        const int h = ty * TILE + ly0 + k * 8;
        const int w = tx * TILE + lx;
        const size_t off = (size_t)h * IMG_W + w;
        const float inv = 1.0f / den[k];
        outn[0 * IMG_HW + off] = num0[k] * inv;
        outn[1 * IMG_HW + off] = num1[k] * inv;
        outn[2 * IMG_HW + off] = num2[k] * inv;
    }
}

extern "C" void kernel_launch(void* const* d_in, const int* in_sizes, int n_in,
                              void* d_out, int out_size, void* d_ws, size_t ws_size,
                              hipStream_t stream)
{
    const float* img    = (const float*)d_in[0];   // (8,3,512,512) fp32
    const float* params = (const float*)d_in[1];   // (8,3) fp32
    float* out          = (float*)d_out;           // (8,3,512,512) fp32

    dim3 grid(IMG_W / TILE, IMG_H / TILE, 8);      // (16,16,8)
    bilateral_kernel<<<grid, 256, 0, stream>>>(img, params, out);
}